// Diffusion_9053791060399
// MI455X (gfx1250) — compile-verified
//
#include <hip/hip_runtime.h>
#include <hip/hip_bf16.h>

// ---------------------------------------------------------------------------
// MI455X (gfx1250) implementation.
//
// Cost model: 14 outputs x [10,128,10000] f32 = ~717 MB of pure streaming
// writes -> HBM-write bound (~31 us @ 23.3 TB/s). The GEMM is 164 MFLOP.
// Strategy: fp32 WMMA (v_wmma_f32_16x16x4_f32) for the 128x10000x64 score
// matmul, then the entire 10-step diffusion-coefficient chain fused in
// registers, streamed out with non-temporal stores.
// ---------------------------------------------------------------------------

typedef float v2f __attribute__((ext_vector_type(2)));
typedef float v8f __attribute__((ext_vector_type(8)));

#define D_DIM 64
#define S_STEPS 10

// ---------------- L2-normalize rows of a [nrows, 64] matrix ----------------
// One wave32 per row: each lane holds a float2 (64 = 32*2), butterfly-reduce
// sum of squares across the wave, scale, store to workspace.
__global__ void l2norm_rows_kernel(const float* __restrict__ src,
                                   const int* __restrict__ gather_idx, // may be null
                                   float* __restrict__ dst,
                                   int nrows) {
    int wave = blockIdx.x * (blockDim.x >> 5) + (threadIdx.x >> 5);
    int lane = threadIdx.x & 31;
    if (wave >= nrows) return;
    int row = gather_idx ? gather_idx[wave] : wave;
    const float2* s = (const float2*)(src + (long long)row * D_DIM);
    float2 v = s[lane];
    float ss = v.x * v.x + v.y * v.y;
#pragma unroll
    for (int m = 16; m >= 1; m >>= 1) ss += __shfl_xor(ss, m, 32);
    float inv = 1.0f / fmaxf(sqrtf(ss), 1e-12f);
    float2* d = (float2*)(dst + (long long)wave * D_DIM);
    d[lane] = make_float2(v.x * inv, v.y * inv);
}

// ---------------- Fused score GEMM + diffusion coefficient chain -----------
// Grid: U/16 blocks of 256 threads (8 waves). Wave w handles B-tile w,
// block x handles U-tile. Each wave computes a 16x16 score tile via 16
// chained V_WMMA_F32_16X16X4_F32, then per element walks the S=10 cumprod
// chain and writes 14 outputs (non-temporal, write-once streaming).
__global__ void __launch_bounds__(256)
diffusion_betas_kernel(const float* __restrict__ u_norm,   // [U,64] normalized users
                       const float* __restrict__ it_norm,  // [B,64] normalized items
                       const int*   __restrict__ inter,    // [B,U]
                       const float* __restrict__ base_betas, // [10]
                       float* __restrict__ out,
                       int U, int B) {
    const int lane = threadIdx.x & 31;
    const int half = lane >> 4;      // 0: lanes 0-15, 1: lanes 16-31
    const int lm   = lane & 15;
    const int bbase = (threadIdx.x >> 5) * 16;  // wave id -> B tile
    const int ubase = blockIdx.x * 16;          // block  -> U tile

    // ---- WMMA f32 16x16x4, K-loop over 64 ----
    // A (16x4 f32): lane<16 holds {K=k0,k0+1} of row M=lm; lane>=16 holds
    // {K=k0+2,k0+3}. B (4x16): same pattern with N=lm, K from rows of u_norm
    // (B[k][n] = u_norm[ubase+n][k]). Both are float2 loads at k0+2*half.
    const float* arow = it_norm + (bbase + lm) * D_DIM;
    const float* brow = u_norm + (ubase + lm) * D_DIM;
    v8f c = {0.f, 0.f, 0.f, 0.f, 0.f, 0.f, 0.f, 0.f};
#pragma unroll
    for (int k0 = 0; k0 < D_DIM; k0 += 4) {
        v2f a = { arow[k0 + 2 * half], arow[k0 + 2 * half + 1] };
        v2f b = { brow[k0 + 2 * half], brow[k0 + 2 * half + 1] };
        c = __builtin_amdgcn_wmma_f32_16x16x4_f32(
            /*neg_a=*/false, a, /*neg_b=*/false, b,
            /*c_mod=*/(short)0, c, /*reuse_a=*/false, /*reuse_b=*/false);
    }

    // Preload the 10 base betas (tiny, L2/const cached).
    float bb[S_STEPS];
#pragma unroll
    for (int t = 0; t < S_STEPS; ++t) bb[t] = base_betas[t];

    const int BU  = B * U;            // 1,280,000
    const int SBU = S_STEPS * BU;     // 12,800,000 (fits int32; 14*SBU < 2^31)
    const int u_idx = ubase + lm;

    // D layout: VGPR j -> M = j + 8*half, N = lm.
#pragma unroll
    for (int j = 0; j < 8; ++j) {
        const int b_idx = bbase + j + 8 * half;
        float s = fminf(fmaxf(c[j], -1.0f), 1.0f);
        if (inter[b_idx * U + u_idx] != 1) s = -s;
        const float gamma = 1.0f - 0.01f * __expf(3.0f * s);

        float acp = 1.0f;
#pragma unroll
        for (int t = 0; t < S_STEPS; ++t) {
            const float beta     = gamma * bb[t];
            const float alpha    = 1.0f - beta;
            const float acp_prev = acp;
            acp *= alpha;

            const float one_m_acp   = 1.0f - acp;
            const float one_m_prev  = 1.0f - acp_prev;
            const float inv_om      = 1.0f / one_m_acp;
            const float som         = sqrtf(one_m_acp);       // sqrt(1-acp)
            const float inv_som     = 1.0f / som;
            const float sacp        = sqrtf(acp);
            const float racp        = rsqrtf(acp);            // sqrt(1/acp)
            const float sprev       = sqrtf(acp_prev);
            const float somprev     = sqrtf(one_m_prev);
            // look-ahead: acp_next[t] = acp[t+1] = acp * alpha[t+1]
            float acp_next = 0.0f;
            if (t < S_STEPS - 1) acp_next = acp * (1.0f - gamma * bb[t + 1]);

            const float o0  = rsqrtf(alpha);                  // sqrt_recip_alphas
            const float o2  = som;                            // sqrt_one_minus_acp
            const float o6  = __logf(one_m_acp);              // log_one_minus_acp
            const float o8  = som * racp;                     // sqrt(1/acp - 1)
            const float o9  = beta * sprev * inv_om;          // posterior_mean_coef1
            const float o10 = one_m_prev * sqrtf(alpha) * inv_om; // posterior_mean_coef2
            const float o11 = somprev * inv_som;              // fast_posterior_coef2
            const float o12 = sacp * somprev * inv_som;       // fast_posterior_coef3
            const float o13 = beta * one_m_prev * inv_om;     // posterior_variance

            const int base = t * BU + b_idx * U + u_idx;
            float* p = out + base;
            __builtin_nontemporal_store(o0,       p + 0  * SBU);
            __builtin_nontemporal_store(acp,      p + 1  * SBU);
            __builtin_nontemporal_store(o2,       p + 2  * SBU);
            __builtin_nontemporal_store(acp_prev, p + 3  * SBU);
            __builtin_nontemporal_store(acp_next, p + 4  * SBU);
            __builtin_nontemporal_store(sacp,     p + 5  * SBU);
            __builtin_nontemporal_store(o6,       p + 6  * SBU);
            __builtin_nontemporal_store(racp,     p + 7  * SBU);
            __builtin_nontemporal_store(o8,       p + 8  * SBU);
            __builtin_nontemporal_store(o9,       p + 9  * SBU);
            __builtin_nontemporal_store(o10,      p + 10 * SBU);
            __builtin_nontemporal_store(o11,      p + 11 * SBU);
            __builtin_nontemporal_store(o12,      p + 12 * SBU);
            __builtin_nontemporal_store(o13,      p + 13 * SBU);
        }
    }
}

extern "C" void kernel_launch(void* const* d_in, const int* in_sizes, int n_in,
                              void* d_out, int out_size, void* d_ws, size_t ws_size,
                              hipStream_t stream) {
    const float* user_emb   = (const float*)d_in[0];  // [U,64]
    const float* item_emb   = (const float*)d_in[1];  // [N,64]
    const int*   iids       = (const int*)  d_in[2];  // [B]
    const int*   inter      = (const int*)  d_in[3];  // [B,U]
    const float* base_betas = (const float*)d_in[4];  // [10]

    const int U = in_sizes[0] / D_DIM;   // 10000
    const int B = in_sizes[2];           // 128

    float* u_norm  = (float*)d_ws;                       // U*64 floats (2.56 MB)
    float* it_norm = u_norm + (size_t)U * D_DIM;         // B*64 floats

    // Normalize users: one wave per row, 8 waves per block.
    {
        int waves = U;
        int blocks = (waves * 32 + 255) / 256;
        l2norm_rows_kernel<<<blocks, 256, 0, stream>>>(user_emb, nullptr, u_norm, U);
    }
    // Gather + normalize items.
    {
        int waves = B;
        int blocks = (waves * 32 + 255) / 256;
        l2norm_rows_kernel<<<blocks, 256, 0, stream>>>(item_emb, iids, it_norm, B);
    }
    // Fused WMMA score + diffusion chain. U/16 blocks x (B/16 waves).
    {
        dim3 grid(U / 16);
        dim3 block((B / 16) * 32);   // 256 threads = 8 waves
        diffusion_betas_kernel<<<grid, block, 0, stream>>>(
            u_norm, it_norm, inter, base_betas, (float*)d_out, U, B);
    }
}